// Retrieval_loss_44212393345714
// MI455X (gfx1250) — compile-verified
//
#include <hip/hip_runtime.h>
#include <hip/hip_bf16.h>

// Problem constants (from reference setup_inputs)
#define BS 16
#define NPTS 2048
#define MPTS 2048
#define KSEG 32
#define DTOK 128

#define MIN_SUPPORT 20.0f
#define MAX_BEAR_ERROR 20.0f
#define AVG_ERROR 0.05f
#define WEIGHT 1.0f

typedef __attribute__((ext_vector_type(2))) float v2f;
typedef __attribute__((ext_vector_type(8))) float v8f;

// ---------------------------------------------------------------------------
// Kernel 1: rigid transform chain -> A-vectors [x0,x1,x2,|x|^2] per point.
// (the homogeneous "1" for the K=4 WMMA A column is materialized in-register)
// ---------------------------------------------------------------------------
__global__ __launch_bounds__(256) void k_transform(
    const float* __restrict__ pc_partial,   // [BS,N,3]
    const float* __restrict__ p_rot, const float* __restrict__ p_t,
    const float* __restrict__ f_rot, const float* __restrict__ f_t,
    const float* __restrict__ tgt_rot, const float* __restrict__ tgt_t,
    const float* __restrict__ src_rot, const float* __restrict__ src_t,
    float4* __restrict__ xA)                // [BS*N]
{
    int idx = blockIdx.x * blockDim.x + threadIdx.x;
    if (idx >= BS * NPTS) return;
    int b = idx / NPTS;

    const float* pr = p_rot   + b * 9;  const float* pt = p_t   + b * 3;
    const float* tr = tgt_rot + b * 9;  const float* tt = tgt_t + b * 3;
    const float* sr = src_rot + b * 9;  const float* st = src_t + b * 3;
    const float* fr = f_rot   + b * 9;  const float* ft = f_t   + b * 3;

    float p0 = pc_partial[idx * 3 + 0];
    float p1 = pc_partial[idx * 3 + 1];
    float p2 = pc_partial[idx * 3 + 2];

    // z = p @ p_rot + p_t
    float z0 = p0 * pr[0] + p1 * pr[3] + p2 * pr[6] + pt[0];
    float z1 = p0 * pr[1] + p1 * pr[4] + p2 * pr[7] + pt[1];
    float z2 = p0 * pr[2] + p1 * pr[5] + p2 * pr[8] + pt[2];
    // z = (z - tgt_t) @ tgt_rot^T
    float a0 = z0 - tt[0], a1 = z1 - tt[1], a2 = z2 - tt[2];
    float w0 = a0 * tr[0] + a1 * tr[1] + a2 * tr[2];
    float w1 = a0 * tr[3] + a1 * tr[4] + a2 * tr[5];
    float w2 = a0 * tr[6] + a1 * tr[7] + a2 * tr[8];
    // z = z @ src_rot + src_t
    float u0 = w0 * sr[0] + w1 * sr[3] + w2 * sr[6] + st[0];
    float u1 = w0 * sr[1] + w1 * sr[4] + w2 * sr[7] + st[1];
    float u2 = w0 * sr[2] + w1 * sr[5] + w2 * sr[8] + st[2];
    // x = (z - f_t) @ f_rot^T
    float c0 = u0 - ft[0], c1 = u1 - ft[1], c2 = u2 - ft[2];
    float x0 = c0 * fr[0] + c1 * fr[1] + c2 * fr[2];
    float x1 = c0 * fr[3] + c1 * fr[4] + c2 * fr[5];
    float x2 = c0 * fr[6] + c1 * fr[7] + c2 * fr[8];

    float4 o;
    o.x = x0; o.y = x1; o.z = x2;
    o.w = x0 * x0 + x1 * x1 + x2 * x2;      // |x|^2, folded in after the min
    xA[idx] = o;
}

// ---------------------------------------------------------------------------
// Kernel 2: deformed [BS,3,M] -> B-vectors [-2y0,-2y1,-2y2,|y|^2]
// ---------------------------------------------------------------------------
__global__ __launch_bounds__(256) void k_prepB(
    const float* __restrict__ deformed,     // [BS,3,M]
    float4* __restrict__ yB)                // [BS*M]
{
    int idx = blockIdx.x * blockDim.x + threadIdx.x;
    if (idx >= BS * MPTS) return;
    int b = idx / MPTS;
    int m = idx - b * MPTS;
    float y0 = deformed[((size_t)b * 3 + 0) * MPTS + m];
    float y1 = deformed[((size_t)b * 3 + 1) * MPTS + m];
    float y2 = deformed[((size_t)b * 3 + 2) * MPTS + m];
    float4 o;
    o.x = -2.0f * y0; o.y = -2.0f * y1; o.z = -2.0f * y2;
    o.w = y0 * y0 + y1 * y1 + y2 * y2;
    yB[idx] = o;
}

// ---------------------------------------------------------------------------
// Kernel 3: WMMA chamfer-min.
//   S = A(16x4) @ B(4x16) per tile via V_WMMA_F32_16X16X4_F32,
//   running elementwise min over 128 column tiles, then cross-lane min
//   within each 16-lane half (C/D layout: lanes = cols, VGPRr = row r / r+8).
// One wave per 16-row tile of N; 8 waves per block. No prefetch: the per-batch
// B stream is 32 KB, re-read by 16 blocks -> L2 (192 MB) resident.
// ---------------------------------------------------------------------------
#define CHAM_WAVES 8
__global__ __launch_bounds__(CHAM_WAVES * 32) void k_chamfer(
    const float4* __restrict__ xA,          // [BS*N]
    const float4* __restrict__ yB,          // [BS*M]
    float* __restrict__ cham)               // [BS*N]
{
    const int lane = threadIdx.x & 31;
    const int wave = threadIdx.x >> 5;
    const int half = lane >> 4;             // 0: K=0,1  1: K=2,3 (A/B VGPR layout)
    const int sub  = lane & 15;             // row index (A) / col index (B) in tile

    const int blocksPerBatch = NPTS / (16 * CHAM_WAVES);    // 16
    const int b  = blockIdx.x / blocksPerBatch;
    const int n0 = (blockIdx.x % blocksPerBatch) * (16 * CHAM_WAVES) + wave * 16;

    // A tile: rows n0..n0+15, homogeneous 4th component = 1.0
    float4 xa = xA[(size_t)b * NPTS + n0 + sub];
    v2f A;
    A[0] = half ? xa.z : xa.x;              // K=2 : K=0
    A[1] = half ? 1.0f : xa.y;              // K=3 : K=1

    v8f acc;
#pragma unroll
    for (int i = 0; i < 8; ++i) acc[i] = 3.402823466e+38f;

    const float4* yb_base = yB + (size_t)b * MPTS;

#pragma unroll 4
    for (int mt = 0; mt < MPTS / 16; ++mt) {
        float4 yb = yb_base[mt * 16 + sub];
        v2f B;
        B[0] = half ? yb.z : yb.x;          // K=2 : K=0
        B[1] = half ? yb.w : yb.y;          // K=3 : K=1
        v8f c = {};
        v8f d = __builtin_amdgcn_wmma_f32_16x16x4_f32(
            /*neg_a=*/false, A, /*neg_b=*/false, B,
            /*c_mod=*/(short)0, c, /*reuse_a=*/false, /*reuse_b=*/false);
#pragma unroll
        for (int i = 0; i < 8; ++i) acc[i] = fminf(acc[i], d[i]);
    }

    // Row-min across the 16 lanes of each half (xor 1,2,4,8 stays inside half)
    float rowmin[8];
#pragma unroll
    for (int i = 0; i < 8; ++i) {
        float v = acc[i];
        v = fminf(v, __shfl_xor(v, 1, 32));
        v = fminf(v, __shfl_xor(v, 2, 32));
        v = fminf(v, __shfl_xor(v, 4, 32));
        v = fminf(v, __shfl_xor(v, 8, 32));
        rowmin[i] = v;
    }

    // Lanes 0 and 16 write rows [n0 + half*8 .. +7]; add |x|^2, clamp at 0.
    if (sub == 0) {
        int rowbase = n0 + half * 8;
#pragma unroll
        for (int r = 0; r < 8; ++r) {
            float xs = xA[(size_t)b * NPTS + rowbase + r].w;
            cham[(size_t)b * NPTS + rowbase + r] = fmaxf(rowmin[r] + xs, 0.0f);
        }
    }
}

// ---------------------------------------------------------------------------
// Kernel 4: per-batch column reductions over N:
//   seg2def[k] = sum_n seg_p[b,n,k] * cham[b,n] ; sup_p[k] ; sup_f[k]
// One block per batch. Each wave reads the 32 contiguous k-values of one row
// (coalesced 128 B per row per matrix); 8 waves stride over n; LDS reduce.
// ---------------------------------------------------------------------------
__global__ __launch_bounds__(256) void k_seg(
    const float* __restrict__ seg_p,        // [BS,N,K]
    const float* __restrict__ seg_f,        // [BS,N,K]
    const float* __restrict__ cham,         // [BS,N]
    float* __restrict__ s2d,                // [BS*K]
    float* __restrict__ sup_p,              // [BS*K]
    float* __restrict__ sup_f)              // [BS*K]
{
    const int b    = blockIdx.x;
    const int k    = threadIdx.x & 31;      // lane == keypoint index (K=32)
    const int wave = threadIdx.x >> 5;      // 0..7

    float a = 0.0f, s = 0.0f, f = 0.0f;
    for (int n = wave; n < NPTS; n += 8) {
        float ch = cham[(size_t)b * NPTS + n];           // wave-uniform
        float sp = seg_p[((size_t)b * NPTS + n) * KSEG + k];
        float sf = seg_f[((size_t)b * NPTS + n) * KSEG + k];
        a += sp * ch;
        s += sp;
        f += sf;
    }

    __shared__ float r0[8][KSEG], r1[8][KSEG], r2[8][KSEG];
    r0[wave][k] = a; r1[wave][k] = s; r2[wave][k] = f;
    __syncthreads();
    if (wave == 0) {
        float ta = 0.0f, ts = 0.0f, tf = 0.0f;
#pragma unroll
        for (int w = 0; w < 8; ++w) {
            ta += r0[w][k]; ts += r1[w][k]; tf += r2[w][k];
        }
        s2d[b * KSEG + k]   = ta;
        sup_p[b * KSEG + k] = ts;
        sup_f[b * KSEG + k] = tf;
    }
}

// ---------------------------------------------------------------------------
// Kernel 5: final masked mean over BS*K = 512 entries (single block).
// ---------------------------------------------------------------------------
__global__ __launch_bounds__(512) void k_final(
    const float* __restrict__ rtf,          // [BS,K,D]
    const float* __restrict__ rtp,          // [BS,K,D]
    const float* __restrict__ s2d,
    const float* __restrict__ sup_p,
    const float* __restrict__ sup_f,
    float* __restrict__ out)
{
    int tid = threadIdx.x;                  // == b*K + k
    float sp_raw = sup_p[tid];
    float sf_raw = sup_f[tid];
    bool valid = (sp_raw >= MIN_SUPPORT) && (sf_raw >= MIN_SUPPORT);
    float sp = fmaxf(sp_raw, 1.0f);
    float sf = fmaxf(sf_raw, 1.0f);

    float t = s2d[tid] / sp / AVG_ERROR;
    float rel = 1.0f / (1.0f + __expf(-t));

    const float* pf = rtf + (size_t)tid * DTOK;
    const float* pp = rtp + (size_t)tid * DTOK;
    float isf = 1.0f / sf, isp = 1.0f / sp;
    float rdis = 0.0f;
#pragma unroll 4
    for (int d = 0; d < DTOK; ++d) {
        float diff = pf[d] * isf - pp[d] * isp;
        rdis += diff * diff;
    }
    float loss = rdis - rel;
    loss = loss * loss;
    bool m = valid && (loss <= MAX_BEAR_ERROR);
    float v = m ? loss : 0.0f;
    float c = m ? 1.0f : 0.0f;

    __shared__ float ssum[512], scnt[512];
    ssum[tid] = v; scnt[tid] = c;
    __syncthreads();
    for (int stride = 256; stride > 0; stride >>= 1) {
        if (tid < stride) {
            ssum[tid] += ssum[tid + stride];
            scnt[tid] += scnt[tid + stride];
        }
        __syncthreads();
    }
    if (tid == 0) out[0] = ssum[0] / (scnt[0] + 1.0f) * WEIGHT;
}

// ---------------------------------------------------------------------------
extern "C" void kernel_launch(void* const* d_in, const int* in_sizes, int n_in,
                              void* d_out, int out_size, void* d_ws, size_t ws_size,
                              hipStream_t stream) {
    const float* r_tokens_full    = (const float*)d_in[0];
    const float* r_tokens_partial = (const float*)d_in[1];
    const float* pc_seg_partial   = (const float*)d_in[2];
    const float* pc_seg_full      = (const float*)d_in[3];
    const float* pc_partial       = (const float*)d_in[4];
    const float* deformed         = (const float*)d_in[5];
    const float* p_rot            = (const float*)d_in[6];
    const float* p_t              = (const float*)d_in[7];
    const float* f_rot            = (const float*)d_in[8];
    const float* f_t              = (const float*)d_in[9];
    const float* tgt_rand_rot     = (const float*)d_in[10];
    const float* tgt_rand_t       = (const float*)d_in[11];
    const float* src_rand_rot     = (const float*)d_in[12];
    const float* src_rand_t       = (const float*)d_in[13];

    // Workspace layout (floats); total = 296448 floats = 1.16 MB
    float* xA   = (float*)d_ws;                 // BS*N*4
    float* yB   = xA   + BS * NPTS * 4;         // BS*M*4
    float* cham = yB   + BS * MPTS * 4;         // BS*N
    float* s2d  = cham + BS * NPTS;             // BS*K
    float* supp = s2d  + BS * KSEG;             // BS*K
    float* supf = supp + BS * KSEG;             // BS*K

    k_transform<<<(BS * NPTS + 255) / 256, 256, 0, stream>>>(
        pc_partial, p_rot, p_t, f_rot, f_t,
        tgt_rand_rot, tgt_rand_t, src_rand_rot, src_rand_t,
        (float4*)xA);

    k_prepB<<<(BS * MPTS + 255) / 256, 256, 0, stream>>>(
        deformed, (float4*)yB);

    k_chamfer<<<BS * (NPTS / (16 * CHAM_WAVES)), CHAM_WAVES * 32, 0, stream>>>(
        (const float4*)xA, (const float4*)yB, cham);

    k_seg<<<BS, 256, 0, stream>>>(
        pc_seg_partial, pc_seg_full, cham, s2d, supp, supf);

    k_final<<<1, 512, 0, stream>>>(
        r_tokens_full, r_tokens_partial, s2d, supp, supf, (float*)d_out);
}